// GNN_14431090114715
// MI455X (gfx1250) — compile-verified
//
#include <hip/hip_runtime.h>
#include <hip/hip_bf16.h>

typedef __attribute__((ext_vector_type(2))) float v2f;
typedef __attribute__((ext_vector_type(8))) float v8f;

#define NN 50000
#define NE 800000
#define DF 96
#define NG 100
#define NC 8

// ---------------- zero ----------------
__global__ void GNN_zero_kernel(float* __restrict__ p, int n) {
    int i = blockIdx.x * blockDim.x + threadIdx.x;
    if (i < n) p[i] = 0.0f;
}

// ---------------- edge scatter-add aggregation ----------------
// tid -> (edge, 4-float chunk). 24 chunks of 4 floats cover DF=96.
// agg[dst] += feat[src]; atomics land in L2 (agg buffer = 19.2MB << 192MB L2).
__global__ void GNN_scatter_kernel(const float* __restrict__ feat,
                                   const int* __restrict__ ei,
                                   float* __restrict__ agg) {
    int tid = blockIdx.x * blockDim.x + threadIdx.x;
    int e = tid / 24;
    int c = (tid % 24) * 4;
    int src = ei[e];
    int dst = ei[NE + e];
    const float4 v = *(const float4*)(feat + (size_t)src * DF + c);
    float* o = agg + (size_t)dst * DF + c;
    atomicAdd(o + 0, v.x);
    atomicAdd(o + 1, v.y);
    atomicAdd(o + 2, v.z);
    atomicAdd(o + 3, v.w);
}

// ---------------- fused dual-GEMM + bias (+ReLU) via f32 WMMA ----------------
// out = act(A1 @ B1 + A2 @ B2 + bias),  A:[NN,96] B:[96,96] row-major.
// One wave32 per 16x16 output tile, V_WMMA_F32_16X16X4_F32, K stepped by 4.
// A frag (16x4 f32): lanes 0-15 -> M=0..15 K={k,k+1}; lanes 16-31 -> K={k+2,k+3}.
// B frag (4x16 f32): lanes 0-15 -> N=0..15 K={k,k+1}; lanes 16-31 -> K={k+2,k+3}.
// C/D: VGPR v, lanes 0-15 -> M=v N=lane; lanes 16-31 -> M=v+8 N=lane-16.
template <bool RELU>
__global__ __launch_bounds__(192)
void GNN_gemm_kernel(const float* __restrict__ A1,
                     const float* __restrict__ A2,
                     const float* __restrict__ B1,
                     const float* __restrict__ B2,
                     const float* __restrict__ bias,
                     float* __restrict__ out) {
    const int lane = threadIdx.x & 31;
    const int wid  = threadIdx.x >> 5;    // 0..5 -> N tile
    const int m0 = blockIdx.x * 16;       // 3125 blocks, 50000 = 3125*16 exactly
    const int n0 = wid * 16;              // 96 = 6*16 exactly
    const int r  = lane & 15;
    const int kh = lane >> 4;             // K half-select (0 or 1)

    const float* a1p = A1 + (size_t)(m0 + r) * DF + 2 * kh;
    const float* a2p = A2 + (size_t)(m0 + r) * DF + 2 * kh;
    const float* b1p = B1 + (size_t)(2 * kh) * DF + n0 + r;
    const float* b2p = B2 + (size_t)(2 * kh) * DF + n0 + r;

    v8f acc = {};
#pragma unroll
    for (int k = 0; k < DF; k += 4) {
        v2f a = *(const v2f*)(a1p + k);                       // contiguous K pair
        v2f b;
        b.x = b1p[(size_t)k * DF];
        b.y = b1p[(size_t)(k + 1) * DF];
        acc = __builtin_amdgcn_wmma_f32_16x16x4_f32(false, a, false, b,
                                                    (short)0, acc, false, false);
        v2f a2 = *(const v2f*)(a2p + k);
        v2f b2;
        b2.x = b2p[(size_t)k * DF];
        b2.y = b2p[(size_t)(k + 1) * DF];
        acc = __builtin_amdgcn_wmma_f32_16x16x4_f32(false, a2, false, b2,
                                                    (short)0, acc, false, false);
    }

    const float bv = bias[n0 + r];
#pragma unroll
    for (int v = 0; v < 8; ++v) {
        float val = acc[v] + bv;
        if (RELU) val = fmaxf(val, 0.0f);
        out[(size_t)(m0 + v + 8 * kh) * DF + n0 + r] = val;
    }
}

// ---------------- global mean pool (scatter stage) ----------------
__global__ void GNN_pool_kernel(const float* __restrict__ h,
                                const int* __restrict__ batch,
                                float* __restrict__ psum,
                                float* __restrict__ pcnt) {
    int tid = blockIdx.x * blockDim.x + threadIdx.x;
    int node = tid / DF;
    int f = tid % DF;
    int g = batch[node];
    atomicAdd(psum + (size_t)g * DF + f, h[(size_t)node * DF + f]);
    if (f == 0) atomicAdd(pcnt + g, 1.0f);
}

// ---------------- classifier head: (psum/cnt) @ Wl + bl ----------------
__global__ void GNN_head_kernel(const float* __restrict__ psum,
                                const float* __restrict__ pcnt,
                                const float* __restrict__ Wl,
                                const float* __restrict__ bl,
                                float* __restrict__ out) {
    int tid = blockIdx.x * blockDim.x + threadIdx.x;
    if (tid >= NG * NC) return;
    int g = tid / NC;
    int c = tid % NC;
    float inv = 1.0f / fmaxf(pcnt[g], 1.0f);
    float s = 0.0f;
#pragma unroll 8
    for (int k = 0; k < DF; ++k)
        s += psum[(size_t)g * DF + k] * Wl[k * NC + c];
    out[tid] = s * inv + bl[c];
}

extern "C" void kernel_launch(void* const* d_in, const int* in_sizes, int n_in,
                              void* d_out, int out_size, void* d_ws, size_t ws_size,
                              hipStream_t stream) {
    const float* x    = (const float*)d_in[0];
    const int*   ei   = (const int*)d_in[1];
    // d_in[2] = edge_attr, unused by GraphConv
    const int*   batch= (const int*)d_in[3];
    const float* Wn1  = (const float*)d_in[4];
    const float* Ws1  = (const float*)d_in[5];
    const float* b1   = (const float*)d_in[6];
    const float* Wn2  = (const float*)d_in[7];
    const float* Ws2  = (const float*)d_in[8];
    const float* b2   = (const float*)d_in[9];
    const float* Wn3  = (const float*)d_in[10];
    const float* Ws3  = (const float*)d_in[11];
    const float* b3   = (const float*)d_in[12];
    const float* Wl   = (const float*)d_in[13];
    const float* bl   = (const float*)d_in[14];
    float* out = (float*)d_out;

    // workspace layout (floats): agg | hA | hB | psum | pcnt
    float* agg  = (float*)d_ws;
    float* hA   = agg + (size_t)NN * DF;
    float* hB   = hA  + (size_t)NN * DF;
    float* psum = hB  + (size_t)NN * DF;
    float* pcnt = psum + (size_t)NG * DF;

    const int nFeat = NN * DF;                  // 4.8M
    const int zBlocks = (nFeat + 255) / 256;    // 18750
    const int sBlocks = (NE * 24) / 256;        // 75000 (exact)
    const int gBlocks = NN / 16;                // 3125 (exact)

    // ---- layer 1: h = relu(agg(x) @ Wn1 + x @ Ws1 + b1) ----
    GNN_zero_kernel<<<zBlocks, 256, 0, stream>>>(agg, nFeat);
    GNN_scatter_kernel<<<sBlocks, 256, 0, stream>>>(x, ei, agg);
    GNN_gemm_kernel<true><<<gBlocks, 192, 0, stream>>>(agg, x, Wn1, Ws1, b1, hA);

    // ---- layer 2 ----
    GNN_zero_kernel<<<zBlocks, 256, 0, stream>>>(agg, nFeat);
    GNN_scatter_kernel<<<sBlocks, 256, 0, stream>>>(hA, ei, agg);
    GNN_gemm_kernel<true><<<gBlocks, 192, 0, stream>>>(agg, hA, Wn2, Ws2, b2, hB);

    // ---- layer 3 (no relu) ----
    GNN_zero_kernel<<<zBlocks, 256, 0, stream>>>(agg, nFeat);
    GNN_scatter_kernel<<<sBlocks, 256, 0, stream>>>(hB, ei, agg);
    GNN_gemm_kernel<false><<<gBlocks, 192, 0, stream>>>(agg, hB, Wn3, Ws3, b3, hA);

    // ---- global mean pool + head ----
    GNN_zero_kernel<<<(NG * DF + NG + 255) / 256, 256, 0, stream>>>(psum, NG * DF + NG);
    GNN_pool_kernel<<<(NN * DF) / 256, 256, 0, stream>>>(hA, batch, psum, pcnt);
    GNN_head_kernel<<<(NG * NC + 255) / 256, 256, 0, stream>>>(psum, pcnt, Wl, bl, out);
}